// Encoder_42588895707231
// MI455X (gfx1250) — compile-verified
//
#include <hip/hip_runtime.h>

// D = hidden dim = 128, fixed by the reference.
#define DDIM 128
#define BLOCK_THREADS 256   // 8 wave32 waves
#define ROWS_PER_BLOCK 128  // 16 rows per wave

typedef __attribute__((ext_vector_type(2))) float v2f;
typedef __attribute__((ext_vector_type(4))) float v4f;
typedef __attribute__((ext_vector_type(8))) float v8f;

// ---------------------------------------------------------------------------
// Zero-fill d_out (harness poisons it with 0xAA; scatter-sum needs zeros).
// ---------------------------------------------------------------------------
__global__ __launch_bounds__(BLOCK_THREADS) void zero_out_kernel(float* __restrict__ out,
                                                                 int n4) {
  int i = blockIdx.x * blockDim.x + threadIdx.x;
  if (i < n4) {
    v4f z = {};
    ((v4f*)out)[i] = z;
  }
}

// ---------------------------------------------------------------------------
// Stage W (128x128) into LDS with K-pair interleaving so a B fragment for
// V_WMMA_F32_16X16X4_F32 is one ds_load_b64:
//   lds_w[2*((k>>1)*128 + n) + (k&1)] = W[k][n]
// Also stage the bias vector.
// ---------------------------------------------------------------------------
__device__ __forceinline__ void stage_w_bias(const float* __restrict__ w,
                                             const float* __restrict__ bias,
                                             float* lds_w, float* lds_b) {
  int tid = threadIdx.x;
  for (int idx = tid; idx < DDIM * DDIM; idx += BLOCK_THREADS) {
    int k = idx >> 7;
    int n = idx & 127;
    lds_w[(((k >> 1) * DDIM + n) << 1) + (k & 1)] = w[idx];
  }
  if (tid < DDIM) lds_b[tid] = bias[tid];
  __syncthreads();
}

// ---------------------------------------------------------------------------
// One wave computes a 16x128 tile of (arow_gathered[16x128] @ W[128x128])
// using V_WMMA_F32_16X16X4_F32 (exact f32).
//
// A layout (16x4 f32): lanes 0-15 -> M=0..15, VGPR0/1 = K0/K1;
//                      lanes 16-31 -> M=0..15, VGPR0/1 = K2/K3.
//   -> lane loads a contiguous f32 pair: arow[m][kc*4 + half*2 .. +1]
// B layout (4x16 f32): lanes 0-15 hold K0/K1 rows, lanes 16-31 hold K2/K3,
//   N = lane%16 -> one b64 from the interleaved LDS image.
// ---------------------------------------------------------------------------
__device__ __forceinline__ void wmma_tile_16x128(const float* __restrict__ arow,
                                                 const float* __restrict__ lds_w,
                                                 int half, int m, v8f acc[8]) {
  for (int kc = 0; kc < DDIM / 4; ++kc) {
    v2f a = *(const v2f*)(arow + kc * 4 + half * 2);
    int kp = kc * 2 + half;  // k-pair index
    const float* bp = lds_w + 2 * (kp * DDIM + m);
#pragma unroll
    for (int t = 0; t < 8; ++t) {
      v2f b = *(const v2f*)(bp + t * 32);
      acc[t] = __builtin_amdgcn_wmma_f32_16x16x4_f32(
          /*neg_a=*/false, a, /*neg_b=*/false, b,
          /*c_mod=*/(short)0, acc[t], /*reuse_a=*/false, /*reuse_b=*/false);
    }
  }
}

// ---------------------------------------------------------------------------
// Node pass: node_h[r,:] = relu(emb[node_tokens[r],:] @ W1 + b1)
// ---------------------------------------------------------------------------
__global__ __launch_bounds__(BLOCK_THREADS) void node_mlp_kernel(
    const float* __restrict__ emb, const float* __restrict__ w1,
    const float* __restrict__ b1, const int* __restrict__ tokens,
    float* __restrict__ node_h, int n_nodes) {
  __shared__ float lds_w[DDIM * DDIM];
  __shared__ float lds_b[DDIM];
  stage_w_bias(w1, b1, lds_w, lds_b);

  const int lane = threadIdx.x & 31;
  const int wave = threadIdx.x >> 5;
  const int half = lane >> 4;
  const int m = lane & 15;
  const int rowBase = blockIdx.x * ROWS_PER_BLOCK + wave * 16;

  // Clamp the gather row so EXEC stays all-1s through the WMMAs.
  int myRow = rowBase + m;
  int tokRow = myRow < n_nodes ? myRow : n_nodes - 1;
  int tok = tokens[tokRow];
  const float* arow = emb + (long)tok * DDIM;

  v8f zero = {};
  v8f acc[8];
#pragma unroll
  for (int t = 0; t < 8; ++t) acc[t] = zero;

  wmma_tile_16x128(arow, lds_w, half, m, acc);

  // C/D layout: element j of acc[t] is row (j + 8*half), col (t*16 + m).
#pragma unroll
  for (int t = 0; t < 8; ++t) {
    float bias = lds_b[t * 16 + m];
#pragma unroll
    for (int j = 0; j < 8; ++j) {
      int r = rowBase + half * 8 + j;
      if (r < n_nodes) {
        float v = acc[t][j] + bias;
        v = v > 0.f ? v : 0.f;
        node_h[(long)r * DDIM + t * 16 + m] = v;
      }
    }
  }
}

// ---------------------------------------------------------------------------
// Edge pass (fused): edge_h = relu(emb[edge_tokens]@W2+b2);
// msg = node_h[src] * edge_h; atomically scatter-add into out[dst].
// ---------------------------------------------------------------------------
__global__ __launch_bounds__(BLOCK_THREADS) void edge_msg_kernel(
    const float* __restrict__ emb, const float* __restrict__ w2,
    const float* __restrict__ b2, const int* __restrict__ tokens,
    const int* __restrict__ src, const int* __restrict__ dst,
    const float* __restrict__ node_h, float* __restrict__ out, int n_edges) {
  __shared__ float lds_w[DDIM * DDIM];
  __shared__ float lds_b[DDIM];
  stage_w_bias(w2, b2, lds_w, lds_b);

  const int lane = threadIdx.x & 31;
  const int wave = threadIdx.x >> 5;
  const int half = lane >> 4;
  const int m = lane & 15;
  const int rowBase = blockIdx.x * ROWS_PER_BLOCK + wave * 16;

  int myRow = rowBase + m;
  int tokRow = myRow < n_edges ? myRow : n_edges - 1;
  int tok = tokens[tokRow];
  const float* arow = emb + (long)tok * DDIM;

  v8f zero = {};
  v8f acc[8];
#pragma unroll
  for (int t = 0; t < 8; ++t) acc[t] = zero;

  wmma_tile_16x128(arow, lds_w, half, m, acc);

  float biasv[8];
#pragma unroll
  for (int t = 0; t < 8; ++t) biasv[t] = lds_b[t * 16 + m];

#pragma unroll
  for (int j = 0; j < 8; ++j) {
    int r = rowBase + half * 8 + j;  // edge index held in acc[*][j] by this lane
    if (r < n_edges) {
      int s = src[r];
      int d = dst[r];
      const float* nh = node_h + (long)s * DDIM + m;
      float* op = out + (long)d * DDIM + m;
#pragma unroll
      for (int t = 0; t < 8; ++t) {
        float v = acc[t][j] + biasv[t];
        v = v > 0.f ? v : 0.f;
        v *= nh[t * 16];
        atomicAdd(op + t * 16, v);
      }
    }
  }
}

// ---------------------------------------------------------------------------
extern "C" void kernel_launch(void* const* d_in, const int* in_sizes, int n_in,
                              void* d_out, int out_size, void* d_ws, size_t ws_size,
                              hipStream_t stream) {
  const float* emb = (const float*)d_in[0];
  const float* w1  = (const float*)d_in[1];
  const float* b1  = (const float*)d_in[2];
  const float* w2  = (const float*)d_in[3];
  const float* b2  = (const float*)d_in[4];
  const int* node_tokens = (const int*)d_in[5];
  const int* edge_tokens = (const int*)d_in[6];
  const int* src = (const int*)d_in[7];
  const int* dst = (const int*)d_in[8];
  float* out = (float*)d_out;

  const int n_nodes = in_sizes[5];
  const int n_edges = in_sizes[6];

  float* node_h = (float*)d_ws;  // n_nodes * 128 floats (~25.6 MB)

  int n4 = out_size / 4;  // out_size = n_nodes * 128, divisible by 4
  zero_out_kernel<<<(n4 + BLOCK_THREADS - 1) / BLOCK_THREADS, BLOCK_THREADS, 0, stream>>>(out, n4);

  int nodeBlocks = (n_nodes + ROWS_PER_BLOCK - 1) / ROWS_PER_BLOCK;
  node_mlp_kernel<<<nodeBlocks, BLOCK_THREADS, 0, stream>>>(emb, w1, b1, node_tokens,
                                                            node_h, n_nodes);

  int edgeBlocks = (n_edges + ROWS_PER_BLOCK - 1) / ROWS_PER_BLOCK;
  edge_msg_kernel<<<edgeBlocks, BLOCK_THREADS, 0, stream>>>(emb, w2, b2, edge_tokens,
                                                            src, dst, node_h, out, n_edges);
}